// SparseGraphEncoder_2594160246838
// MI455X (gfx1250) — compile-verified
//
#include <hip/hip_runtime.h>

#define NNODES 100000
#define NEDGES 1600000
#define NGRAPH 256
#define NHEADS 4

typedef float v2f __attribute__((ext_vector_type(2)));
typedef float v8f __attribute__((ext_vector_type(8)));

static __device__ __forceinline__ float leaky02(float v) {
    return v > 0.0f ? v : 0.2f * v;
}

static __device__ __forceinline__ void atomicMaxF(float* addr, float val) {
    int* ia = (int*)addr;
    int old = __float_as_int(*addr);
    while (__int_as_float(old) < val) {
        int assumed = old;
        old = atomicCAS(ia, assumed, __float_as_int(val));
        if (old == assumed) break;
    }
}

// ---------------- elementwise helpers ----------------

__global__ void k_fill(float* p, float v, long long n) {
    long long i = (long long)blockIdx.x * blockDim.x + threadIdx.x;
    if (i < n) p[i] = v;
}

__global__ void k_degree(const int* __restrict__ col, float* deg, int e) {
    int i = blockIdx.x * blockDim.x + threadIdx.x;
    if (i < e) atomicAdd(&deg[col[i]], 1.0f);
}

__global__ void k_rsqrt(float* p, int n) {
    int i = blockIdx.x * blockDim.x + threadIdx.x;
    if (i < n) p[i] = rsqrtf(p[i]);
}

// ---------------- dense GEMM: Y[n,dout] = X[n,din] @ W[din,dout] ----------------
// One wave32 per 16x16 output tile using V_WMMA_F32_16X16X4_F32 (full f32 path).
// A 16x4 layout: lanes 0-15 -> K=0,1 ; lanes 16-31 -> K=2,3 (M = lane&15)
// B 4x16 layout: lanes 0-15 -> K=0,1 ; lanes 16-31 -> K=2,3 (N = lane&15)
// C/D: VGPR v -> M = v + 8*(lane>=16), N = lane&15

__global__ void k_gemm_wmma(const float* __restrict__ X, const float* __restrict__ W,
                            float* __restrict__ Y, int n, int din, int dout) {
    const int lane  = threadIdx.x & 31;
    const int wave  = (blockIdx.x * blockDim.x + threadIdx.x) >> 5;
    const int tilesN = dout >> 4;
    const int tileM = wave / tilesN;
    const int tileN = wave - tileM * tilesN;
    if (tileM * 16 >= n) return;  // uniform per wave; EXEC stays all-ones in live waves

    const int m  = lane & 15;
    const int kh = (lane >> 4) << 1;  // 0 or 2
    const float* xr = X + (size_t)(tileM * 16 + m) * din;
    const float* wc = W + tileN * 16 + m;

    v8f acc = {};
    for (int k0 = 0; k0 < din; k0 += 4) {
        v2f a, b;
        a.x = xr[k0 + kh];
        a.y = xr[k0 + kh + 1];
        b.x = wc[(size_t)(k0 + kh) * dout];
        b.y = wc[(size_t)(k0 + kh + 1) * dout];
        acc = __builtin_amdgcn_wmma_f32_16x16x4_f32(
            /*neg_a=*/false, a, /*neg_b=*/false, b,
            /*c_mod=*/(short)0, acc, /*reuse_a=*/false, /*reuse_b=*/false);
    }

    const int mb = (lane >> 4) << 3;  // 0 or 8
    float* yr = Y + (size_t)(tileM * 16 + mb) * dout + tileN * 16 + m;
#pragma unroll
    for (int v = 0; v < 8; ++v) yr[(size_t)v * dout] = acc[v];
}

// ---------------- GCN normalized aggregation (vec4 channels) ----------------
// out[col] += h[row] * dis[row]*dis[col]; entries >= E are self loops.
// One thread per (edge, 4-channel chunk): b128 gather + 4 f32 atomics in one line.

__global__ void k_gcn_agg(const float* __restrict__ h, const int* __restrict__ row,
                          const int* __restrict__ col, const float* __restrict__ dis,
                          float* out, int dout, long long total4) {
    long long idx = (long long)blockIdx.x * blockDim.x + threadIdx.x;
    if (idx >= total4) return;
    const int q = dout >> 2;
    int c4 = (int)(idx % q) << 2;
    long long e = idx / q;
    int j, i;
    if (e < NEDGES) { j = row[e]; i = col[e]; } else { j = i = (int)(e - NEDGES); }
    float w = dis[j] * dis[i];
    const float4 hv = *(const float4*)(h + (size_t)j * dout + c4);
    float* op = out + (size_t)i * dout + c4;
    atomicAdd(op + 0, hv.x * w);
    atomicAdd(op + 1, hv.y * w);
    atomicAdd(op + 2, hv.z * w);
    atomicAdd(op + 3, hv.w * w);
}

// ---------------- BatchNorm (training-mode, biased var, eps=1e-5) ----------------

__global__ void k_bn_stats(const float* __restrict__ y, float* stat, int dout, long long total4) {
    __shared__ float s0[128];
    __shared__ float s1[128];
    for (int c = threadIdx.x; c < dout; c += blockDim.x) { s0[c] = 0.0f; s1[c] = 0.0f; }
    __syncthreads();
    const int q = dout >> 2;
    for (long long idx = (long long)blockIdx.x * blockDim.x + threadIdx.x; idx < total4;
         idx += (long long)gridDim.x * blockDim.x) {
        int c4 = (int)(idx % q) << 2;
        long long r = idx / q;
        float4 v = *(const float4*)(y + (size_t)r * dout + c4);
        atomicAdd(&s0[c4 + 0], v.x); atomicAdd(&s1[c4 + 0], v.x * v.x);
        atomicAdd(&s0[c4 + 1], v.y); atomicAdd(&s1[c4 + 1], v.y * v.y);
        atomicAdd(&s0[c4 + 2], v.z); atomicAdd(&s1[c4 + 2], v.z * v.z);
        atomicAdd(&s0[c4 + 3], v.w); atomicAdd(&s1[c4 + 3], v.w * v.w);
    }
    __syncthreads();
    for (int c = threadIdx.x; c < dout; c += blockDim.x) {
        atomicAdd(&stat[c], s0[c]);
        atomicAdd(&stat[dout + c], s1[c]);
    }
}

__global__ void k_bn_apply(const float* __restrict__ yin, float* yout,
                           const float* __restrict__ stat, const float* __restrict__ g,
                           const float* __restrict__ b, int dout, long long total4, int relu) {
    long long idx = (long long)blockIdx.x * blockDim.x + threadIdx.x;
    if (idx >= total4) return;
    const int q = dout >> 2;
    int c4 = (int)(idx % q) << 2;
    long long r = idx / q;
    const float inv_n = 1.0f / (float)NNODES;
    float4 v = *(const float4*)(yin + (size_t)r * dout + c4);
    float vv[4] = {v.x, v.y, v.z, v.w};
    float o[4];
#pragma unroll
    for (int t = 0; t < 4; ++t) {
        int c = c4 + t;
        float mu  = stat[c] * inv_n;
        float var = stat[dout + c] * inv_n - mu * mu;
        float rr = rsqrtf(var + 1e-5f);
        float x = (vv[t] - mu) * rr * g[c] + b[c];
        if (relu) x = leaky02(x);
        o[t] = x;
    }
    *(float4*)(yout + (size_t)r * dout + c4) = make_float4(o[0], o[1], o[2], o[3]);
}

// ---------------- GAT ----------------

__global__ void k_gat_coef(const float* __restrict__ h, const float* __restrict__ a_src,
                           const float* __restrict__ a_dst, float* asrc, float* adst,
                           int C) {
    int idx = blockIdx.x * blockDim.x + threadIdx.x;
    if (idx >= NNODES * NHEADS) return;
    int node = idx / NHEADS;
    int hd   = idx - node * NHEADS;
    const float* hp = h + (size_t)node * NHEADS * C + hd * C;
    const float* as = a_src + hd * C;
    const float* ad = a_dst + hd * C;
    float s = 0.0f, d = 0.0f;
    for (int c = 0; c < C; ++c) { float v = hp[c]; s += v * as[c]; d += v * ad[c]; }
    asrc[idx] = s;
    adst[idx] = d;
}

__global__ void k_gat_edge1(const float* __restrict__ asrc, const float* __restrict__ adst,
                            const int* __restrict__ row, const int* __restrict__ col,
                            float* ebuf, float* segm, long long total) {
    long long idx = (long long)blockIdx.x * blockDim.x + threadIdx.x;
    if (idx >= total) return;
    int hd = (int)(idx % NHEADS);
    long long e = idx / NHEADS;
    int j, i;
    if (e < NEDGES) { j = row[e]; i = col[e]; } else { j = i = (int)(e - NEDGES); }
    float v = leaky02(asrc[j * NHEADS + hd] + adst[i * NHEADS + hd]);
    ebuf[idx] = v;
    atomicMaxF(&segm[i * NHEADS + hd], v);
}

__global__ void k_gat_edge2(float* ebuf, const float* __restrict__ segm, float* segs,
                            const int* __restrict__ row, const int* __restrict__ col,
                            long long total) {
    long long idx = (long long)blockIdx.x * blockDim.x + threadIdx.x;
    if (idx >= total) return;
    int hd = (int)(idx % NHEADS);
    long long e = idx / NHEADS;
    int i;
    if (e < NEDGES) { i = col[e]; } else { i = (int)(e - NEDGES); }
    float ex = expf(ebuf[idx] - segm[i * NHEADS + hd]);
    ebuf[idx] = ex;
    atomicAdd(&segs[i * NHEADS + hd], ex);
}

// One thread per (edge, 4-channel chunk). C (= d/4) is a multiple of 4, so a
// 4-channel chunk never crosses a head boundary -> single alpha per chunk.
__global__ void k_gat_edge3(const float* __restrict__ h, const float* __restrict__ ebuf,
                            const float* __restrict__ segs, const int* __restrict__ row,
                            const int* __restrict__ col, float* out, int d, int C,
                            long long total4) {
    long long idx = (long long)blockIdx.x * blockDim.x + threadIdx.x;
    if (idx >= total4) return;
    const int q = d >> 2;
    int c4 = (int)(idx % q) << 2;
    long long e = idx / q;
    int j, i;
    if (e < NEDGES) { j = row[e]; i = col[e]; } else { j = i = (int)(e - NEDGES); }
    int hd = c4 / C;
    float alpha = ebuf[e * NHEADS + hd] / segs[i * NHEADS + hd];
    const float4 hv = *(const float4*)(h + (size_t)j * d + c4);
    float* op = out + (size_t)i * d + c4;
    atomicAdd(op + 0, hv.x * alpha);
    atomicAdd(op + 1, hv.y * alpha);
    atomicAdd(op + 2, hv.z * alpha);
    atomicAdd(op + 3, hv.w * alpha);
}

__global__ void k_gat_final(float* out, const float* __restrict__ bias,
                            const float* __restrict__ xres, int d, long long total4) {
    long long idx = (long long)blockIdx.x * blockDim.x + threadIdx.x;
    if (idx >= total4) return;
    const int q = d >> 2;
    int c4 = (int)(idx % q) << 2;
    long long r = idx / q;
    size_t off = (size_t)r * d + c4;
    float4 o = *(const float4*)(out + off);
    float4 x = *(const float4*)(xres + off);
    float4 bb = *(const float4*)(bias + c4);
    o.x = leaky02(o.x + bb.x + x.x);
    o.y = leaky02(o.y + bb.y + x.y);
    o.z = leaky02(o.z + bb.z + x.z);
    o.w = leaky02(o.w + bb.w + x.w);
    *(float4*)(out + off) = o;
}

// ---------------- global mean pool ----------------

__global__ void k_count(const int* __restrict__ batch, float* cnt) {
    int i = blockIdx.x * blockDim.x + threadIdx.x;
    if (i < NNODES) atomicAdd(&cnt[batch[i]], 1.0f);
}

__global__ void k_pool(const float* __restrict__ x, const int* __restrict__ batch,
                       float* pool, int d, long long total4) {
    long long idx = (long long)blockIdx.x * blockDim.x + threadIdx.x;
    if (idx >= total4) return;
    const int q = d >> 2;
    int c4 = (int)(idx % q) << 2;
    int nidx = (int)(idx / q);
    const float4 v = *(const float4*)(x + (size_t)nidx * d + c4);
    float* op = pool + (size_t)batch[nidx] * d + c4;
    atomicAdd(op + 0, v.x);
    atomicAdd(op + 1, v.y);
    atomicAdd(op + 2, v.z);
    atomicAdd(op + 3, v.w);
}

__global__ void k_mean(const float* __restrict__ pool, const float* __restrict__ cnt,
                       float* out, int d) {
    int idx = blockIdx.x * blockDim.x + threadIdx.x;
    if (idx >= NGRAPH * d) return;
    int g = idx / d;
    out[idx] = pool[idx] / fmaxf(cnt[g], 1.0f);
}

// ---------------- launch ----------------

static inline unsigned blocksFor(long long total, int bs) {
    return (unsigned)((total + bs - 1) / bs);
}

extern "C" void kernel_launch(void* const* d_in, const int* in_sizes, int n_in,
                              void* d_out, int out_size, void* d_ws, size_t ws_size,
                              hipStream_t stream) {
    const float* x_in  = (const float*)d_in[0];
    const int*   eidx  = (const int*)d_in[1];
    const int*   batch = (const int*)d_in[2];
    const float* gcn_w[4] = {(const float*)d_in[3],  (const float*)d_in[7],
                             (const float*)d_in[11], (const float*)d_in[15]};
    const float* bn_g[4]  = {(const float*)d_in[5],  (const float*)d_in[9],
                             (const float*)d_in[13], (const float*)d_in[17]};
    const float* bn_b[4]  = {(const float*)d_in[6],  (const float*)d_in[10],
                             (const float*)d_in[14], (const float*)d_in[18]};
    const float* gat_w[2]  = {(const float*)d_in[19], (const float*)d_in[23]};
    const float* gat_as[2] = {(const float*)d_in[20], (const float*)d_in[24]};
    const float* gat_ad[2] = {(const float*)d_in[21], (const float*)d_in[25]};
    const float* gat_b[2]  = {(const float*)d_in[22], (const float*)d_in[26]};

    const int* row = eidx;           // source nodes j
    const int* col = eidx + NEDGES;  // target nodes i

    // workspace layout (floats)
    float* ws = (float*)d_ws;
    const size_t NB = (size_t)NNODES * 128;
    float* b0   = ws;
    float* b1   = ws + NB;
    float* b2   = ws + 2 * NB;
    float* dis  = ws + 3 * NB;                       // N
    float* asrc = dis + NNODES;                      // N*H
    float* adst = asrc + (size_t)NNODES * NHEADS;    // N*H
    float* segm = adst + (size_t)NNODES * NHEADS;    // N*H
    float* segs = segm + (size_t)NNODES * NHEADS;    // N*H
    float* stat = segs + (size_t)NNODES * NHEADS;    // 256
    float* pool = stat + 256;                        // G*64
    float* cnt  = pool + (size_t)NGRAPH * 64;        // G
    float* ebuf = cnt + NGRAPH;                      // (E+N)*H

    const long long TE = (long long)NEDGES + NNODES;

    // symmetric normalization: dis = rsqrt(1 + in-degree at col)
    k_fill<<<blocksFor(NNODES, 256), 256, 0, stream>>>(dis, 1.0f, NNODES);
    k_degree<<<blocksFor(NEDGES, 256), 256, 0, stream>>>(col, dis, NEDGES);
    k_rsqrt<<<blocksFor(NNODES, 256), 256, 0, stream>>>(dis, NNODES);

    auto gcn = [&](const float* X, int din, int dout, int li, float* H, float* AGG, int relu) {
        int waves = (NNODES / 16) * (dout / 16);
        k_gemm_wmma<<<(waves + 7) / 8, 256, 0, stream>>>(X, gcn_w[li], H, NNODES, din, dout);
        hipMemsetAsync(AGG, 0, (size_t)NNODES * dout * sizeof(float), stream);
        long long tot4 = TE * (dout / 4);
        k_gcn_agg<<<blocksFor(tot4, 256), 256, 0, stream>>>(H, row, col, dis, AGG, dout, tot4);
        // gcn bias cancels inside BatchNorm -> skipped
        hipMemsetAsync(stat, 0, 2 * dout * sizeof(float), stream);
        long long tn4 = (long long)NNODES * (dout / 4);
        k_bn_stats<<<1024, 256, 0, stream>>>(AGG, stat, dout, tn4);
        k_bn_apply<<<blocksFor(tn4, 256), 256, 0, stream>>>(AGG, AGG, stat, bn_g[li], bn_b[li],
                                                            dout, tn4, relu);
    };

    auto gat = [&](const float* X, int d, int gi, float* H, float* OUT) {
        int C = d / NHEADS;
        int waves = (NNODES / 16) * (d / 16);
        k_gemm_wmma<<<(waves + 7) / 8, 256, 0, stream>>>(X, gat_w[gi], H, NNODES, d, d);
        int tnh = NNODES * NHEADS;
        k_gat_coef<<<blocksFor(tnh, 256), 256, 0, stream>>>(H, gat_as[gi], gat_ad[gi],
                                                            asrc, adst, C);
        k_fill<<<blocksFor(tnh, 256), 256, 0, stream>>>(segm, -3.0e38f, tnh);
        hipMemsetAsync(segs, 0, (size_t)tnh * sizeof(float), stream);
        long long teh = TE * NHEADS;
        k_gat_edge1<<<blocksFor(teh, 256), 256, 0, stream>>>(asrc, adst, row, col, ebuf, segm, teh);
        k_gat_edge2<<<blocksFor(teh, 256), 256, 0, stream>>>(ebuf, segm, segs, row, col, teh);
        hipMemsetAsync(OUT, 0, (size_t)NNODES * d * sizeof(float), stream);
        long long ted4 = TE * (d / 4);
        k_gat_edge3<<<blocksFor(ted4, 256), 256, 0, stream>>>(H, ebuf, segs, row, col, OUT,
                                                              d, C, ted4);
        long long tnd4 = (long long)NNODES * (d / 4);
        k_gat_final<<<blocksFor(tnd4, 256), 256, 0, stream>>>(OUT, gat_b[gi], X, d, tnd4);
    };

    // layer 0: GCN(16->64) + BN, then GAT residual + leaky_relu
    gcn(x_in, 16, 64, 0, b0, b1, 0);   // X := b1
    gat(b1, 64, 0, b0, b2);            // X := b2
    // layer 1: GCN(64->128) + BN + leaky_relu
    gcn(b2, 64, 128, 1, b0, b1, 1);    // X := b1
    // layer 2: GCN(128->128) + BN, then GAT residual + leaky_relu
    gcn(b1, 128, 128, 2, b0, b2, 0);   // X := b2
    gat(b2, 128, 1, b0, b1);           // X := b1
    // layer 3: GCN(128->64) + BN (no activation)
    gcn(b1, 128, 64, 3, b0, b2, 0);    // X := b2

    // global mean pool -> [G, 64]
    hipMemsetAsync(pool, 0, (size_t)(NGRAPH * 64 + NGRAPH) * sizeof(float), stream);
    k_count<<<blocksFor(NNODES, 256), 256, 0, stream>>>(batch, cnt);
    long long tp4 = (long long)NNODES * (64 / 4);
    k_pool<<<blocksFor(tp4, 256), 256, 0, stream>>>(b2, batch, pool, 64, tp4);
    k_mean<<<blocksFor(NGRAPH * 64, 256), 256, 0, stream>>>(pool, cnt, (float*)d_out, 64);
}